// BatchedPolicy_62801011802285
// MI455X (gfx1250) — compile-verified
//
#include <hip/hip_runtime.h>
#include <math.h>

// ---------------------------------------------------------------------------
// CV-QNN forward for MI455X (gfx1250): LDS-resident state, f32 WMMA GEMMs,
// TDM (tensor_load_to_lds) for gate-matrix streaming into LDS.
// B=256 batch, D=4 modes, CUT=10, LAYERS=4, P=48 weights/layer.
// ---------------------------------------------------------------------------

#define NB     256
#define ND     4
#define NC     10
#define NLAY   4
#define PW     48
#define KST    25       // K = 100 = 25 * 4  (f32 WMMA K=4, no K padding)
#define ABUF   11200    // pitch-100 panel alloc: covers reads up to row 111

struct c32 { float x, y; };

typedef float v2f __attribute__((ext_vector_type(2)));
typedef float v8f __attribute__((ext_vector_type(8)));
typedef unsigned int v4u __attribute__((ext_vector_type(4)));
typedef int v4i __attribute__((ext_vector_type(4)));
typedef int v8i __attribute__((ext_vector_type(8)));

__device__ __forceinline__ c32 cmul(c32 a, c32 b) {
  return c32{ a.x*b.x - a.y*b.y, a.x*b.y + a.y*b.x };
}

__device__ __forceinline__ int stride4(int m) {
  return m == 0 ? 1000 : (m == 1 ? 100 : (m == 2 ? 10 : 1));
}

// workspace layout in units of c32 (8 bytes)
constexpr int WS_PSI0   = 0;                       // 16  (10 used)
constexpr int WS_PSI    = 16;                      // 256*4*10 = 10240
constexpr int WS_SMALLU = 16 + NB*ND*NC;           // 32 * 100
constexpr int WS_BIGU   = WS_SMALLU + 32*100;      // 48 * 10000

// evolve-kernel dynamic LDS layout (units of c32)
constexpr int EV_STATE  = 0;        // 10016
constexpr int EV_APAD   = 10016;    // 11200
constexpr int EV_USTG   = 21216;    // 11200 (gate matrix, TDM destination)
constexpr int EV_SBUF   = 32416;    // 128
constexpr unsigned EV_USTG_BYTES = (unsigned)EV_USTG * 8u;

// ---------------------------------------------------------------------------
// TDM: DMA one contiguous 80KB gate matrix (10000 x 8B) global -> LDS.
// Issued by wave 0 only (TDM ignores EXEC; avoid 8 duplicate DMAs).
// Descriptor per CDNA5 ISA.
// This toolchain's builtin is the 6-arg flavor (g0, g1, g2, g3, g4, cpol).
// ---------------------------------------------------------------------------
__device__ __forceinline__ void tdm_load_gate(const c32* __restrict__ Ug,
                                              unsigned lds_byte_off)
{
  const unsigned long long ga = (unsigned long long)(uintptr_t)Ug;
  v4u g0;
  g0.x = 1u;                                            // count=1, user D#
  g0.y = lds_byte_off;                                  // lds_addr
  g0.z = (unsigned)(ga & 0xFFFFFFFFu);                  // global_addr[31:0]
  g0.w = (unsigned)((ga >> 32) & 0x01FFFFFFu) | (2u << 30);  // [56:32] | type=2
  v8i g1;
  g1[0] = (int)(3u << 16);          // wg_mask=0, data_size=3 (8B)
  g1[1] = (int)(10000u << 16);      // tensor_dim0[15:0] in bits [63:48]
  g1[2] = (int)(1u << 16);          // tensor_dim0 hi=0 | tensor_dim1=1
  g1[3] = (int)(10000u << 16);      // tensor_dim1 hi=0 | tile_dim0=10000
  g1[4] = (int)1;                   // tile_dim1=1 | tile_dim2=0
  g1[5] = (int)10000;               // tensor_dim0_stride lo
  g1[6] = 0;                        // stride hi | tensor_dim1_stride lo
  g1[7] = 0;
  v4i gz4 = {0, 0, 0, 0};
  v8i gz8 = {0, 0, 0, 0, 0, 0, 0, 0};
  __builtin_amdgcn_tensor_load_to_lds(g0, g1, gz4, gz4, gz8, 0);
}

// ---------------------------------------------------------------------------
// 10x10 complex expm, Horner (12 terms) + 6 squarings of M/64.
// ---------------------------------------------------------------------------
__device__ c32* expm10(c32* X, c32* Y, c32* Z, int tid)
{
  __syncthreads();
  if (tid < 100) {
    X[tid].x *= 0.015625f; X[tid].y *= 0.015625f;   // 2^-6
    int i = tid / 10, j = tid % 10;
    Y[tid] = c32{ (i == j) ? 1.f : 0.f, 0.f };
  }
  __syncthreads();
  c32* py = Y; c32* pz = Z;
  for (int t = 12; t >= 1; --t) {
    if (tid < 100) {
      int i = tid / 10, j = tid % 10;
      float re = 0.f, im = 0.f;
      #pragma unroll
      for (int k = 0; k < 10; ++k) {
        c32 xa = X[i*10 + k], yb = py[k*10 + j];
        re += xa.x*yb.x - xa.y*yb.y;
        im += xa.x*yb.y + xa.y*yb.x;
      }
      float inv = 1.f / (float)t;
      pz[tid] = c32{ re*inv + ((i == j) ? 1.f : 0.f), im*inv };
    }
    __syncthreads();
    c32* tp = py; py = pz; pz = tp;
  }
  for (int s = 0; s < 6; ++s) {
    if (tid < 100) {
      int i = tid / 10, j = tid % 10;
      float re = 0.f, im = 0.f;
      #pragma unroll
      for (int k = 0; k < 10; ++k) {
        c32 xa = py[i*10 + k], yb = py[k*10 + j];
        re += xa.x*yb.x - xa.y*yb.y;
        im += xa.x*yb.y + xa.y*yb.x;
      }
      pz[tid] = c32{ re, im };
    }
    __syncthreads();
    c32* tp = py; py = pz; pz = tp;
  }
  return py;   // == Y (18 swaps)
}

// ---------------------------------------------------------------------------
// Kernel 1: small 10x10 gates (S0/psi0, squeezers, layer displacements).
// ---------------------------------------------------------------------------
__global__ __launch_bounds__(128)
void small_gates_kernel(const float* __restrict__ w, c32* __restrict__ ws)
{
  __shared__ c32 X[100], Y[100], Z[100];
  const int blk = blockIdx.x, tid = threadIdx.x;
  if (tid < 100) {
    int i = tid / 10, j = tid % 10;
    c32 g = {0.f, 0.f};
    if (blk == 0) {
      if (j == i + 2)      { float s1 = sqrtf((float)((i+1)*(i+2))); g = c32{0.f, -0.25f*s1}; }
      else if (i == j + 2) { float s1 = sqrtf((float)((j+1)*(j+2))); g = c32{0.f, -0.25f*s1}; }
    } else if (blk <= 16) {
      int id = blk - 1, l = id / 4, m = id % 4;
      float r = w[l*PW + 16 + m];
      if (j == i + 2)      { float s1 = sqrtf((float)((i+1)*(i+2))); g = c32{ 0.5f*r*s1, 0.f}; }
      else if (i == j + 2) { float s1 = sqrtf((float)((j+1)*(j+2))); g = c32{-0.5f*r*s1, 0.f}; }
    } else {
      int id = blk - 17, l = id / 4, m = id % 4;
      float rd = w[l*PW + 36 + m], phd = w[l*PW + 40 + m];
      float sa, ca; __sincosf(phd, &sa, &ca);
      if (i == j + 1)      { float s1 = sqrtf((float)i); g = c32{ rd*ca*s1, rd*sa*s1}; }
      else if (j == i + 1) { float s1 = sqrtf((float)j); g = c32{-rd*ca*s1, rd*sa*s1}; }
    }
    X[tid] = g;
  }
  c32* R = expm10(X, Y, Z, tid);
  if (blk == 0) {
    if (tid < 10) ws[WS_PSI0 + tid] = R[tid*10 + 0];
  } else if (blk <= 16) {
    if (tid < 100) ws[WS_SMALLU + (blk - 1)*100 + tid] = R[tid];
  } else {
    if (tid < 100) ws[WS_SMALLU + (16 + blk - 17)*100 + tid] = R[tid];
  }
}

// ---------------------------------------------------------------------------
// Kernel 2: per-(b,m) input displacement: psi = expm(r(adag - a)) @ psi0.
// ---------------------------------------------------------------------------
__global__ __launch_bounds__(128)
void psi_kernel(const float* __restrict__ inputs, c32* __restrict__ ws)
{
  __shared__ c32 X[100], Y[100], Z[100];
  __shared__ c32 p0[10];
  const int id = blockIdx.x, tid = threadIdx.x;
  const float r = inputs[id];
  if (tid < 100) {
    int i = tid / 10, j = tid % 10;
    float v = 0.f;
    if (i == j + 1)      v =  r * sqrtf((float)i);
    else if (j == i + 1) v = -r * sqrtf((float)j);
    X[tid] = c32{v, 0.f};
  }
  if (tid < 10) p0[tid] = ws[WS_PSI0 + tid];
  c32* R = expm10(X, Y, Z, tid);
  if (tid < 10) {
    float re = 0.f, im = 0.f;
    #pragma unroll
    for (int n = 0; n < 10; ++n) {
      c32 u = R[tid*10 + n], v = p0[n];
      re += u.x*v.x - u.y*v.y;
      im += u.x*v.y + u.y*v.x;
    }
    ws[WS_PSI + id*10 + tid] = c32{re, im};
  }
}

// ---------------------------------------------------------------------------
// Block-cooperative complex GEMM, M=N=100 (7x7 tiles of 16, masked), K=100
// exactly (25 WMMA k-steps). f32 WMMA 16x16x4, 8 waves x 49 tiles.
//   BT=true : B given column-major (Bt[c*100+k])  -> contiguous frag pairs
//   BT=false: B given row-major    (B[k*100+c])
// Complex: Or = ArBr - AiBi (Bi negated in VALU; f32 WMMA NEG is C-only),
//          Oi = ArBi + AiBr.  Out-of-range tiles produce garbage that the
//          masked writeback discards (WMMA raises no FP exceptions).
// ---------------------------------------------------------------------------
template <bool BT, typename F>
__device__ __forceinline__ void cgemm100(const c32* __restrict__ A,
                                         const c32* __restrict__ B, F wb)
{
  const int lane = threadIdx.x & 31;
  const int wave = threadIdx.x >> 5;
  const int half = lane >> 4;        // K-half select (A/B fragment layout)
  const int l16  = lane & 15;
  for (int tile = wave; tile < 49; tile += 8) {
    const int mt = tile / 7, nt = tile % 7;
    v8f Or = {}; v8f Oi = {};
    const int arow = mt*16 + l16;
    const int bcol = nt*16 + l16;
    for (int kk = 0; kk < KST; ++kk) {
      const int k0 = kk*4 + half*2;
      const c32 a0 = A[arow*100 + k0];
      const c32 a1 = A[arow*100 + k0 + 1];
      c32 b0, b1;
      if (BT) { b0 = B[bcol*100 + k0]; b1 = B[bcol*100 + k0 + 1]; }
      else    { b0 = B[k0*100 + bcol]; b1 = B[(k0 + 1)*100 + bcol]; }
      v2f Ar = { a0.x,  a1.x };
      v2f Ai = { a0.y,  a1.y };
      v2f Br = { b0.x,  b1.x };
      v2f Bi = { b0.y,  b1.y };
      v2f Bn = { -b0.y, -b1.y };
      Or = __builtin_amdgcn_wmma_f32_16x16x4_f32(false, Ar, false, Br, (short)0, Or, false, false);
      Or = __builtin_amdgcn_wmma_f32_16x16x4_f32(false, Ai, false, Bn, (short)0, Or, false, false);
      Oi = __builtin_amdgcn_wmma_f32_16x16x4_f32(false, Ar, false, Bi, (short)0, Oi, false, false);
      Oi = __builtin_amdgcn_wmma_f32_16x16x4_f32(false, Ai, false, Br, (short)0, Oi, false, false);
    }
    #pragma unroll
    for (int e = 0; e < 8; ++e)
      wb(mt*16 + e + half*8, nt*16 + l16, Or[e], Oi[e]);
  }
}

// ---------------------------------------------------------------------------
// Kernel 3: 48 two-mode gates U = expm(t(e^{ip}Aab - e^{-ip}Aab^T)), 100x100,
// WMMA GEMM Horner (12 terms) + 6 squarings. LDS: 3 x 11200 c32 (~269 KB).
// ---------------------------------------------------------------------------
__global__ __launch_bounds__(256)
void bigU_kernel(const float* __restrict__ w, c32* __restrict__ ws)
{
  extern __shared__ char smem[];
  c32* X = (c32*)smem;       // generator (scaled), pitch 100
  c32* Y = X + ABUF;
  c32* Z = Y + ABUF;
  const int g   = blockIdx.x, tid = threadIdx.x;
  const int l   = g / 12, rem = g % 12, half = rem / 6, k = rem % 6;
  const float th = w[l*PW + (half ? 20 : 0) + k];
  const float ph = w[l*PW + (half ? 26 : 6) + k];
  const float SC = 0.015625f;                    // 2^-6 scaling

  for (int i = tid; i < 10000; i += 256) { X[i] = c32{0.f,0.f}; Y[i] = c32{0.f,0.f}; }
  __syncthreads();
  if (tid < 81) {
    int r = tid / 9, q = tid % 9;
    float s1 = sqrtf((float)((r+1)*(q+1)));
    float sp, cp; __sincosf(ph, &sp, &cp);
    float amp = th * s1 * SC;
    int row1 = (r+1)*10 + q;       // Aab[(r+1,q),(r,q+1)]
    int col1 = r*10 + (q+1);
    X[row1*100 + col1] = c32{  amp*cp, amp*sp };  //  t e^{ip}
    X[col1*100 + row1] = c32{ -amp*cp, amp*sp };  // -t e^{-ip}
  }
  if (tid < 100) Y[tid*101] = c32{1.f, 0.f};
  __syncthreads();

  c32* py = Y; c32* pz = Z;
  for (int t = 12; t >= 1; --t) {                // Horner: R = I + (M R)/t
    const float inv = 1.f / (float)t;
    cgemm100<false>(X, py, [&](int row, int col, float re, float im) {
      if (row < 100 && col < 100) {
        float d = (row == col) ? 1.f : 0.f;
        pz[row*100 + col] = c32{ re*inv + d, im*inv };
      }
    });
    __syncthreads();
    c32* tp = py; py = pz; pz = tp;
  }
  for (int s = 0; s < 6; ++s) {                  // squarings
    cgemm100<false>(py, py, [&](int row, int col, float re, float im) {
      if (row < 100 && col < 100)
        pz[row*100 + col] = c32{ re, im };
    });
    __syncthreads();
    c32* tp = py; py = pz; pz = tp;
  }
  c32* dst = ws + WS_BIGU + (size_t)g * 10000;
  for (int i = tid; i < 10000; i += 256)
    dst[i] = py[i];
}

// ---------------------------------------------------------------------------
// Kernel 4: state evolution, one workgroup per batch sample, LDS-resident.
// ---------------------------------------------------------------------------
__device__ void apply_two(c32* state, c32* Apad, c32* Ust,
                          const c32* __restrict__ Ug, int i, int j, int tid)
{
  // Kick off TDM DMA of the 80KB gate matrix into LDS (wave 0 only),
  // overlapped with the LDS gather of the A panel below.
  if (tid < 32) tdm_load_gate(Ug, EV_USTG_BYTES);

  int u = -1, v = -1;
  const int mask = (1 << i) | (1 << j);
  for (int m = 0; m < 4; ++m) if (!((mask >> m) & 1)) { if (u < 0) u = m; else v = m; }
  const int si = stride4(i), sj = stride4(j), su = stride4(u), sv = stride4(v);

  for (int idx = tid; idx < 10000; idx += 256) {
    const int r = idx / 100, q = idx - r*100;
    Apad[idx] = state[(r/10)*su + (r%10)*sv + (q/10)*si + (q%10)*sj];
  }
  if (tid < 32) __builtin_amdgcn_s_wait_tensorcnt((short)0);
  __syncthreads();

  // out[r,p] = sum_q S[r,q] U[p,q]  ->  B^T is U's native row-major layout.
  cgemm100<true>(Apad, Ust, [&](int row, int col, float re, float im) {
    if (row < 100 && col < 100)
      state[(row/10)*su + (row%10)*sv + (col/10)*si + (col%10)*sj] = c32{re, im};
  });
  __syncthreads();
}

__device__ void apply_single(c32* state, c32* sbuf, const c32* __restrict__ Ug,
                             int m, int tid)
{
  if (tid < 100) sbuf[tid] = Ug[tid];
  __syncthreads();
  int o0 = -1, o1 = -1, o2 = -1;
  for (int t = 0; t < 4; ++t) if (t != m) { if (o0 < 0) o0 = t; else if (o1 < 0) o1 = t; else o2 = t; }
  const int s0 = stride4(o0), s1 = stride4(o1), s2 = stride4(o2), sm = stride4(m);
  for (int rr = tid; rr < 1000; rr += 256) {
    const int base = (rr/100)*s0 + ((rr/10)%10)*s1 + (rr%10)*s2;
    c32 vreg[10];
    #pragma unroll
    for (int n = 0; n < 10; ++n) vreg[n] = state[base + n*sm];
    #pragma unroll
    for (int kk = 0; kk < 10; ++kk) {
      float re = 0.f, im = 0.f;
      #pragma unroll
      for (int n = 0; n < 10; ++n) {
        c32 uv = sbuf[kk*10 + n];
        re += uv.x*vreg[n].x - uv.y*vreg[n].y;
        im += uv.x*vreg[n].y + uv.y*vreg[n].x;
      }
      state[base + kk*sm] = c32{re, im};
    }
  }
  __syncthreads();
}

__device__ void diag_phase(c32* state, float v0, float v1, float v2, float v3,
                           bool squared, int tid)
{
  for (int idx = tid; idx < 10000; idx += 256) {
    const int m0 = idx/1000, m1 = (idx/100)%10, m2 = (idx/10)%10, m3 = idx%10;
    float f0 = (float)m0, f1 = (float)m1, f2 = (float)m2, f3 = (float)m3;
    if (squared) { f0 *= f0; f1 *= f1; f2 *= f2; f3 *= f3; }
    const float phi = v0*f0 + v1*f1 + v2*f2 + v3*f3;
    float sp, cp; __sincosf(phi, &sp, &cp);
    c32 sv = state[idx];
    state[idx] = c32{ sv.x*cp - sv.y*sp, sv.x*sp + sv.y*cp };
  }
  __syncthreads();
}

__device__ void interferometer(c32* state, c32* Apad, c32* Ust,
                               const c32* __restrict__ bigU,
                               const float* __restrict__ w,
                               int l, int half, int tid)
{
  const int pi[6] = {0, 0, 0, 1, 1, 2};
  const int pj[6] = {1, 2, 3, 2, 3, 3};
  for (int k = 0; k < 6; ++k) {
    const c32* Ug = bigU + (size_t)(l*12 + half*6 + k) * 10000;
    if (k < 5) __builtin_prefetch(Ug + 10000, 0, 1);   // warm L2 for next gate
    apply_two(state, Apad, Ust, Ug, pi[k], pj[k], tid);
  }
  const float* vp = w + l*PW + (half ? 32 : 12);
  diag_phase(state, vp[0], vp[1], vp[2], vp[3], false, tid);
}

__global__ __launch_bounds__(256)
void evolve_kernel(const float* __restrict__ w, const c32* __restrict__ ws,
                   float* __restrict__ out)
{
  extern __shared__ char smem[];
  c32* state = (c32*)smem + EV_STATE;    // 10016
  c32* Apad  = (c32*)smem + EV_APAD;     // 11200 (pitch-100 A panel)
  c32* Ust   = (c32*)smem + EV_USTG;     // 11200 (gate matrix via TDM)
  c32* sbuf  = (c32*)smem + EV_SBUF;     // 128 (psi vectors / single-mode U)
  float* red = (float*)(sbuf + 128);     // 4 reduction cells
  const int b = blockIdx.x, tid = threadIdx.x;
  const c32* bigU = ws + WS_BIGU;

  if (tid < 40) sbuf[tid] = ws[WS_PSI + b*40 + tid];
  if (tid < 4)  red[tid] = 0.f;
  __syncthreads();
  for (int idx = tid; idx < 10000; idx += 256) {
    const int m0 = idx/1000, m1 = (idx/100)%10, m2 = (idx/10)%10, m3 = idx%10;
    c32 v = sbuf[m0];
    v = cmul(v, sbuf[10 + m1]);
    v = cmul(v, sbuf[20 + m2]);
    v = cmul(v, sbuf[30 + m3]);
    state[idx] = v;
  }
  __syncthreads();

  for (int l = 0; l < NLAY; ++l) {
    interferometer(state, Apad, Ust, bigU, w, l, 0, tid);
    for (int m = 0; m < 4; ++m)
      apply_single(state, sbuf, ws + WS_SMALLU + (l*4 + m)*100, m, tid);
    interferometer(state, Apad, Ust, bigU, w, l, 1, tid);
    for (int m = 0; m < 4; ++m)
      apply_single(state, sbuf, ws + WS_SMALLU + (16 + l*4 + m)*100, m, tid);
    const float* kap = w + l*PW + 44;
    diag_phase(state, kap[0], kap[1], kap[2], kap[3], true, tid);  // Kerr
  }

  float a0 = 0.f, a1 = 0.f, a2 = 0.f, a3 = 0.f;
  for (int idx = tid; idx < 10000; idx += 256) {
    const int m0 = idx/1000, m1 = (idx/100)%10, m2 = (idx/10)%10, m3 = idx%10;
    c32 sv = state[idx];
    const float p = sv.x*sv.x + sv.y*sv.y;
    a0 += m0*p; a1 += m1*p; a2 += m2*p; a3 += m3*p;
  }
  atomicAdd(&red[0], a0); atomicAdd(&red[1], a1);
  atomicAdd(&red[2], a2); atomicAdd(&red[3], a3);
  __syncthreads();
  if (tid < 4) out[b*4 + tid] = red[tid];
}

// ---------------------------------------------------------------------------
extern "C" void kernel_launch(void* const* d_in, const int* in_sizes, int n_in,
                              void* d_out, int out_size, void* d_ws, size_t ws_size,
                              hipStream_t stream)
{
  const float* inputs  = (const float*)d_in[0];   // (256, 4) f32
  const float* weights = (const float*)d_in[1];   // (4, 48) f32
  float* out = (float*)d_out;                     // (256, 4) f32
  c32* ws = (c32*)d_ws;                           // needs ~3.95 MB

  small_gates_kernel<<<33, 128, 0, stream>>>(weights, ws);
  psi_kernel<<<NB*ND, 128, 0, stream>>>(inputs, ws);

  const size_t bigu_smem = (size_t)3 * ABUF * sizeof(c32);              // ~269 KB
  bigU_kernel<<<48, 256, bigu_smem, stream>>>(weights, ws);

  const size_t evo_smem = (size_t)(EV_SBUF + 128) * sizeof(c32) + 64;   // ~260 KB
  evolve_kernel<<<NB, 256, evo_smem, stream>>>(weights, ws, out);
}